// Sparse_linear_67405216743618
// MI455X (gfx1250) — compile-verified
//
#include <hip/hip_runtime.h>

typedef __attribute__((ext_vector_type(2))) float v2f;
typedef __attribute__((ext_vector_type(8))) float v8f;

typedef __attribute__((address_space(1))) int gas_int;   // global addrspace
typedef __attribute__((address_space(3))) int las_int;   // LDS addrspace

#define B_   32
#define C_   64
#define O_   64
#define HW_  32
#define LOC  (HW_ * HW_)

#define AS_STRIDE 68            // pad 64 -> 68 dwords: conflict-free column reads
#define BS_STRIDE 68
#define A_ELEMS   (B_ * AS_STRIDE)
#define B_ELEMS   (O_ * BS_STRIDE)
#define CHUNK_ELEMS (A_ELEMS + B_ELEMS)

#if __has_builtin(__builtin_amdgcn_global_load_async_to_lds_b32)
#define USE_ASYNC 1
#else
#define USE_ASYNC 0
#endif

__device__ __forceinline__ void wait_async_all() {
#if USE_ASYNC
#if __has_builtin(__builtin_amdgcn_s_wait_asynccnt)
    __builtin_amdgcn_s_wait_asynccnt(0);
#else
    asm volatile("s_wait_asynccnt 0x0" ::: "memory");
#endif
#endif
}

// One 4-byte global -> LDS copy (per lane). Async path uses the CDNA5
// GLOBAL_LOAD_ASYNC_TO_LDS_B32 instruction (ASYNCcnt tracked); fallback is a
// plain load+ds_store pair.
__device__ __forceinline__ void copy_g2l(const float* __restrict__ g, float* l) {
#if USE_ASYNC
    __builtin_amdgcn_global_load_async_to_lds_b32(
        (gas_int*)g, (las_int*)l, /*offset=*/0, /*cpol=*/0);
#else
    *l = *g;
#endif
}

// Stage one K-chunk (one (u,v) tap): A = x[:, :, hh, ww] (32x64) and
// B^T = wgt[:, :, h, w, u, v] (64x64) into LDS. 24 dwords per thread.
__device__ __forceinline__ void stage_chunk(
    const float* __restrict__ xp,   // x + hh*HW + ww
    const float* __restrict__ wp,   // wgt + lw*9 + uv
    float* As, float* Bst, int tid)
{
#pragma unroll
    for (int i = 0; i < 8; ++i) {
        int idx = tid + i * 256;          // 0..2047
        int b   = idx >> 6;
        int c   = idx & 63;
        copy_g2l(xp + (size_t)(b * C_ + c) * LOC, &As[b * AS_STRIDE + c]);
    }
#pragma unroll
    for (int i = 0; i < 16; ++i) {
        int idx = tid + i * 256;          // 0..4095
        int o   = idx >> 6;
        int c   = idx & 63;
        copy_g2l(wp + (size_t)(o * C_ + c) * (LOC * 9), &Bst[o * BS_STRIDE + c]);
    }
}

// One workgroup per spatial location (h,w): GEMM M=32(batch) x N=64(O) x
// K=576(C*9) as up-to-9 K-chunks of 64 (one per valid (u,v) tap), double
// buffered. 8 wave32s, each owns one 16x16 (batch x O) tile.
__global__ __launch_bounds__(256) void lc_conv3x3_wmma_f32(
    const float* __restrict__ x,     // [B, C, H, W]
    const float* __restrict__ wgt,   // [O, C, Ho, Wo, 3, 3]
    const float* __restrict__ bias,  // [O, Ho, Wo]
    float* __restrict__ out)         // [B, O, Ho, Wo]
{
    __shared__ float smem[2 * CHUNK_ELEMS];   // 52 KB: double-buffered A + B^T

    const int lw   = blockIdx.x;
    const int h    = lw >> 5;
    const int w    = lw & 31;
    const int tid  = threadIdx.x;
    const int lane = tid & 31;
    const int wave = tid >> 5;
    const int mt   = wave & 1;               // 2 tiles along batch
    const int nt   = wave >> 1;              // 4 tiles along O

    const int l15   = lane & 15;
    const int khalf = (lane >> 4) << 1;      // ISA 16x4 A layout: hi lanes hold K+2,K+3
    const int m     = mt * 16 + l15;
    const int n     = nt * 16 + l15;

    // Valid taps (uniform across block; border taps contribute all-zero A).
    int xoff[9];                             // hh*HW + ww per valid tap
    int uvs[9];
    int nv = 0;
#pragma unroll
    for (int uv = 0; uv < 9; ++uv) {
        int hh = h + uv / 3 - 1;
        int ww = w + uv % 3 - 1;
        if (hh >= 0 && hh < HW_ && ww >= 0 && ww < HW_) {
            xoff[nv] = hh * HW_ + ww;
            uvs[nv]  = uv;
            ++nv;
        }
    }

    // C/D 16x16 f32 layout: VGPR r, lane l -> M = r + 8*(l>>4), N = l&15.
    const float bv = bias[(size_t)n * LOC + lw];
    v8f acc;
#pragma unroll
    for (int i = 0; i < 8; ++i) acc[i] = bv;

    const float* wbase = wgt + (size_t)lw * 9;

    // Prologue: stage chunk 0 into buffer 0.
    stage_chunk(x + xoff[0], wbase + uvs[0], smem, smem + A_ELEMS, tid);

    for (int i = 0; i < nv; ++i) {
        float* As  = smem + (i & 1) * CHUNK_ELEMS;
        float* Bst = As + A_ELEMS;

        wait_async_all();        // this wave's async fills have landed in LDS
        __syncthreads();         // every wave's fills visible; prev compute done

        if (i + 1 < nv) {        // kick off next chunk into the other buffer
            float* As2 = smem + ((i + 1) & 1) * CHUNK_ELEMS;
            stage_chunk(x + xoff[i + 1], wbase + uvs[i + 1], As2, As2 + A_ELEMS, tid);
        }

        // 16 WMMA K-steps of 4 over this chunk.
#pragma unroll
        for (int kk = 0; kk < 64; kk += 4) {
            v2f a = *(const v2f*)&As[m * AS_STRIDE + kk + khalf];
            v2f b = *(const v2f*)&Bst[n * BS_STRIDE + kk + khalf];
            acc = __builtin_amdgcn_wmma_f32_16x16x4_f32(
                /*neg_a=*/false, a, /*neg_b=*/false, b,
                /*c_mod=*/(short)0, acc, /*reuse_a=*/false, /*reuse_b=*/false);
        }
        __syncthreads();
    }

    // Store: M=batch, N=o ; out[b][o][h][w]. Output is L2-resident (8.4 MB),
    // adjacent-w blocks fill out the cachelines.
#pragma unroll
    for (int r = 0; r < 8; ++r) {
        int b = mt * 16 + r + ((lane >> 4) << 3);
        out[(size_t)(b * O_ + n) * LOC + lw] = acc[r];
    }
}

extern "C" void kernel_launch(void* const* d_in, const int* in_sizes, int n_in,
                              void* d_out, int out_size, void* d_ws, size_t ws_size,
                              hipStream_t stream) {
    const float* x    = (const float*)d_in[0];
    const float* wgt  = (const float*)d_in[1];
    const float* bias = (const float*)d_in[2];
    float* out        = (float*)d_out;

    lc_conv3x3_wmma_f32<<<LOC, 256, 0, stream>>>(x, wgt, bias, out);
}